// KVMN_45938970198024
// MI455X (gfx1250) — compile-verified
//
#include <hip/hip_runtime.h>
#include <hip/hip_bf16.h>
#include <math.h>

#define BB 4
#define LL 256
#define KK 256
#define HH 768

typedef float v2f __attribute__((ext_vector_type(2)));
typedef float v8f __attribute__((ext_vector_type(8)));

// ---- helpers for the pipelined WMMA loop (8 x16x16x4 steps = 32 H per chunk)
__device__ __forceinline__ void load_chunk(v2f (&a)[8], v2f (&b)[8],
                                           const float* __restrict__ arow,
                                           const float* __restrict__ brow,
                                           int h) {
    #pragma unroll
    for (int i = 0; i < 8; ++i) {
        a[i] = *(const v2f*)(arow + h + i * 4);
        b[i] = *(const v2f*)(brow + h + i * 4);
    }
}

__device__ __forceinline__ void mm_chunk(v8f& acc,
                                         const v2f (&a)[8], const v2f (&b)[8]) {
    #pragma unroll
    for (int i = 0; i < 8; ++i)
        acc = __builtin_amdgcn_wmma_f32_16x16x4_f32(
            false, a[i], false, b[i], (short)0, acc, false, false);
}

// -------------------------------------------------------------------------
// Kernel 1: scores u[b,l,k] = (1/sqrt(H)) * <hidden[b,l,:], key_emb[key_seq[b,k],:]>
// One workgroup (8 waves) per (b, 16-row l-tile); each wave owns two 16x16
// k-tiles. Double-buffered register chunks keep V_WMMA_F32_16X16X4_F32 fed
// (loads for chunk c+1 issue before the WMMA burst of chunk c).
// -------------------------------------------------------------------------
__global__ __launch_bounds__(256) void kvmn_scores(
    const float* __restrict__ hidden,    // [B,L,H]
    const int*   __restrict__ key_seq,   // [B,K]
    const float* __restrict__ key_emb,   // [KEY_SIZE,H]
    float* __restrict__ u)               // [B,L,K]  (workspace)
{
    const int wave = threadIdx.x >> 5;
    const int lane = threadIdx.x & 31;
    const int b    = blockIdx.x >> 4;    // L/16 = 16 l-tiles per batch
    const int lt   = blockIdx.x & 15;
    const int m    = lane & 15;          // A-row / B-col index for this lane
    const int half = lane >> 4;          // K=0..1 vs K=2..3 of each x4 step

    const float inv_temper = 0.036084391824351615f;  // 1/sqrt(768)

    const float* arow = hidden + ((size_t)(b * LL + lt * 16 + m)) * HH + half * 2;

    for (int kt = wave; kt < 16; kt += 8) {
        const int key = key_seq[b * KK + kt * 16 + m];
        const float* brow = key_emb + (size_t)key * HH + half * 2;

        v2f a0[8], b0[8], a1[8], b1[8];
        v8f acc = {};

        load_chunk(a0, b0, arow, brow, 0);
        // HH = 768 = 12 iterations of 64; last prefetch is predicated out.
        for (int hc = 0; hc < HH; hc += 64) {
            load_chunk(a1, b1, arow, brow, hc + 32);      // always in range
            mm_chunk(acc, a0, b0);
            if (hc + 64 < HH)
                load_chunk(a0, b0, arow, brow, hc + 64);
            mm_chunk(acc, a1, b1);
        }

        // C layout: VGPR r -> row (r + 8*half), col = m
        #pragma unroll
        for (int r = 0; r < 8; ++r) {
            const int lrow = lt * 16 + half * 8 + r;
            u[((size_t)b * LL + lrow) * KK + kt * 16 + m] = acc[r] * inv_temper;
        }
    }
}

// -------------------------------------------------------------------------
// Kernel 2: masked softmax over k for each (b,l) row. One wave32 per row.
// -------------------------------------------------------------------------
__global__ __launch_bounds__(256) void kvmn_softmax(
    const int* __restrict__ key_mask,    // [B,L,K] in {0,1}
    float* __restrict__ u)               // in: u  out: p   [B,L,K]
{
    const int row  = blockIdx.x * 8 + (threadIdx.x >> 5);  // flat b*L + l
    const int lane = threadIdx.x & 31;
    const size_t base = (size_t)row * KK;

    float e[8];
    float s = 0.f;
    #pragma unroll
    for (int i = 0; i < 8; ++i) {
        const int k = lane + i * 32;
        const float mk = (key_mask[base + k] > 0) ? 1.f : 0.f;
        const float v  = expf(u[base + k]) * mk;
        e[i] = v;
        s += v;
    }
    #pragma unroll
    for (int off = 16; off > 0; off >>= 1)
        s += __shfl_xor(s, off, 32);

    const float inv = 1.f / (s + 1e-10f);
    #pragma unroll
    for (int i = 0; i < 8; ++i)
        u[base + lane + i * 32] = e[i] * inv;
}

// -------------------------------------------------------------------------
// Kernel 3: o[b,l,h] = sum_k p[b,l,k] * val_emb[value_matrix[b,l,k]][h] + hidden
// One 256-thread block per (b,l). Deterministic ballot-scan compaction of
// nonzero keys (order = ascending k on every call), then a branchless gather
// loop: 3 coalesced B32 streams per row (h, h+256, h+512), ~50% traffic skip.
// -------------------------------------------------------------------------
__global__ __launch_bounds__(256) void kvmn_output(
    const float* __restrict__ hidden,       // [B,L,H]
    const int*   __restrict__ value_matrix, // [B,L,K]
    const float* __restrict__ val_emb,      // [VAL_SIZE,H]
    const float* __restrict__ p,            // [B,L,K]
    float* __restrict__ out)                // [B,L,H]
{
    __shared__ float cw[KK];
    __shared__ int   cidx[KK];
    __shared__ int   wave_cnt[8];

    const int row = blockIdx.x;             // flat b*L + l
    const int tid = threadIdx.x;
    const int wv  = tid >> 5;
    const int ln  = tid & 31;
    const size_t kbase = (size_t)row * KK;

    // Each thread owns key slot k = tid.
    const float w   = p[kbase + tid];
    const int   idx = value_matrix[kbase + tid];

    const unsigned mask32 = (unsigned)__ballot(w != 0.f);   // wave32: low 32 bits
    if (ln == 0) wave_cnt[wv] = __popc(mask32);
    __syncthreads();

    int my_off = 0, nnz = 0;
    #pragma unroll
    for (int i = 0; i < 8; ++i) {
        const int c = wave_cnt[i];
        if (i < wv) my_off += c;
        nnz += c;
    }
    if (w != 0.f) {
        const unsigned ltmask = (1u << ln) - 1u;
        const int pos = my_off + __popc(mask32 & ltmask);   // ascending-k order
        cw[pos]   = w;
        cidx[pos] = idx;
    }
    __syncthreads();

    float a0 = 0.f, a1 = 0.f, a2 = 0.f;
    for (int j = 0; j < nnz; ++j) {
        const float wj = cw[j];
        const float* vr = val_emb + (size_t)cidx[j] * HH;
        a0 = fmaf(wj, vr[tid],       a0);
        a1 = fmaf(wj, vr[tid + 256], a1);
        a2 = fmaf(wj, vr[tid + 512], a2);
    }

    const size_t hbase = (size_t)row * HH;
    out[hbase + tid]       = a0 + hidden[hbase + tid];
    out[hbase + tid + 256] = a1 + hidden[hbase + tid + 256];
    out[hbase + tid + 512] = a2 + hidden[hbase + tid + 512];
}

// -------------------------------------------------------------------------
extern "C" void kernel_launch(void* const* d_in, const int* in_sizes, int n_in,
                              void* d_out, int out_size, void* d_ws, size_t ws_size,
                              hipStream_t stream) {
    const float* hidden       = (const float*)d_in[0];  // [B,L,H]
    const int*   key_seq      = (const int*)  d_in[1];  // [B,K]
    const int*   value_matrix = (const int*)  d_in[2];  // [B,L,K]
    const int*   key_mask     = (const int*)  d_in[3];  // [B,L,K]
    const float* key_emb      = (const float*)d_in[4];  // [KEY_SIZE,H]
    const float* val_emb      = (const float*)d_in[5];  // [VAL_SIZE,H]
    float*       out          = (float*)d_out;          // [B,L,H]
    float*       u            = (float*)d_ws;           // [B,L,K] = 1 MB scratch

    (void)in_sizes; (void)n_in; (void)out_size; (void)ws_size;

    kvmn_scores<<<BB * (LL / 16), 256, 0, stream>>>(hidden, key_seq, key_emb, u);
    kvmn_softmax<<<(BB * LL) / 8, 256, 0, stream>>>(key_mask, u);
    kvmn_output<<<BB * LL, 256, 0, stream>>>(hidden, value_matrix, val_emb, u, out);
}